// GPUBiasingMultiModelReference_28063316313009
// MI455X (gfx1250) — compile-verified
//
#include <hip/hip_runtime.h>
#include <hip/hip_bf16.h>
#include <stdint.h>
#include <stddef.h>

// ---------------------------------------------------------------------------
// GPUBiasingMultiModel: batched (model,state) row gather + eos patch + alpha
// scale + validity mask.  Pure HBM-streaming problem (~262 MB moved, ~11 us
// roofline @ 23.3 TB/s).  Uses the gfx1250 async global->LDS copy engine with
// a 2-deep double buffer per thread, b128 accesses everywhere, and
// non-temporal stores for the write-once output stream.
// ---------------------------------------------------------------------------

#define AS1 __attribute__((address_space(1)))
#define AS3 __attribute__((address_space(3)))

typedef float v4f __attribute__((ext_vector_type(4)));
typedef int   v4i __attribute__((ext_vector_type(4)));

#define TILE_ELEMS 1024          // 256 threads * 4 elements
#define TILES_PER_BLOCK 4
#define CHUNK (TILE_ELEMS * TILES_PER_BLOCK)   // 4096 elements per block

#if defined(__has_builtin)
#  if __has_builtin(__builtin_amdgcn_global_load_async_to_lds_b128)
#    define HAVE_ASYNC_B128 1
#  endif
#  if __has_builtin(__builtin_amdgcn_s_wait_asynccnt)
#    define HAVE_WAIT_ASYNC 1
#  endif
#endif

__device__ __forceinline__ void async_copy_b128(const void* gsrc, void* lds_dst) {
#ifdef HAVE_ASYNC_B128
  // Probe-discovered signature: (global v4i*, local v4i*, imm offset, imm cpol)
  __builtin_amdgcn_global_load_async_to_lds_b128(
      (AS1 v4i*)gsrc, (AS3 v4i*)lds_dst, 0, 0);
#else
  // ISA: GLOBAL_LOAD_ASYNC_TO_LDS_B128 — VDST = LDS byte address, VADDR = 64b
  // global address (GV mode).  Generic LDS pointers carry the LDS offset in
  // their low 32 bits (aperture layout), so truncation gives the DS address.
  unsigned lds_off = (unsigned)(size_t)lds_dst;
  asm volatile("global_load_async_to_lds_b128 %0, %1, off"
               :: "v"(lds_off), "v"(gsrc)
               : "memory");
#endif
}

__device__ __forceinline__ void wait_async_le2() {
#ifdef HAVE_WAIT_ASYNC
  __builtin_amdgcn_s_wait_asynccnt(2);
#else
  asm volatile("s_wait_asynccnt 2" ::: "memory");
#endif
}

__device__ __forceinline__ void wait_async_le0() {
#ifdef HAVE_WAIT_ASYNC
  __builtin_amdgcn_s_wait_asynccnt(0);
#else
  asm volatile("s_wait_asynccnt 0" ::: "memory");
#endif
}

__global__ __launch_bounds__(256)
void biasing_multimodel_kernel(const int* __restrict__ states,
                               const int* __restrict__ model_ids,
                               const float* __restrict__ score_tables,
                               const int* __restrict__ next_tables,
                               const float* __restrict__ final_weights,
                               const float* __restrict__ alphas,
                               const int* __restrict__ eos_ptr,
                               float* __restrict__ out_scores,
                               int* __restrict__ out_next,
                               int S, int V)
{
  // Per-thread private staging slots; 2-deep double buffer.
  __shared__ __align__(16) float lds_s[2][TILE_ELEMS];
  __shared__ __align__(16) int   lds_n[2][TILE_ELEMS];

  const int b     = blockIdx.y;
  const int tid   = threadIdx.x;
  const int lane4 = tid * 4;

  // Uniform per-block scalars -> SMEM loads.
  const int   mid   = model_ids[b];
  const bool  valid = (mid >= 0);
  const int   m     = valid ? mid : 0;
  const int   s     = states[b];
  const float alpha = alphas[m];
  const int   eos   = eos_ptr[0];
  const float fw    = final_weights[(size_t)m * S + s];

  const size_t row  = ((size_t)m * S + (size_t)s) * (size_t)V;
  const float* srow = score_tables + row;
  const int*   nrow = next_tables  + row;
  float*       orow = out_scores + (size_t)b * (size_t)V;
  int*         xrow = out_next   + (size_t)b * (size_t)V;

  const int chunkBase = blockIdx.x * CHUNK;

  // Clamped load index: keeps async loads in-bounds with full EXEC; only the
  // store is guarded.
  auto loadIdx = [&](int t) -> int {
    int i = chunkBase + t * TILE_ELEMS + lane4;
    return (i > V - 4) ? (V - 4) : i;
  };

  // Prologue: prime buffer 0.
  {
    const int i0 = loadIdx(0);
    async_copy_b128(srow + i0, &lds_s[0][lane4]);
    async_copy_b128(nrow + i0, &lds_n[0][lane4]);
  }

#pragma unroll
  for (int t = 0; t < TILES_PER_BLOCK; ++t) {
    if (t + 1 < TILES_PER_BLOCK) {
      const int i1 = loadIdx(t + 1);
      const int nb = (t + 1) & 1;
      async_copy_b128(srow + i1, &lds_s[nb][lane4]);
      async_copy_b128(nrow + i1, &lds_n[nb][lane4]);
      wait_async_le2();   // async loads complete in order: tile t is resident
    } else {
      wait_async_le0();
    }
    asm volatile("" ::: "memory");

    const int cb = t & 1;
    v4f sc = *reinterpret_cast<const v4f*>(&lds_s[cb][lane4]);   // ds_load_b128
    v4i nx = *reinterpret_cast<const v4i*>(&lds_n[cb][lane4]);

    const int idx = chunkBase + t * TILE_ELEMS + lane4;
    if (idx < V) {
      // eos column patch (applied before alpha scaling, as in the reference).
      const unsigned d = (unsigned)(eos - idx);
      if (d < 4u) sc[d] = fw;

      v4f so;
      v4i xo;
      if (valid) {
        so = sc * alpha;
        xo = nx;
      } else {
        so = (v4f){0.f, 0.f, 0.f, 0.f};
        xo = (v4i){-1, -1, -1, -1};
      }
      // Output is written once and never re-read: NT stores keep the gathered
      // table rows (which may be re-hit across the batch) live in L2.
      __builtin_nontemporal_store(so, reinterpret_cast<v4f*>(orow + idx));
      __builtin_nontemporal_store(xo, reinterpret_cast<v4i*>(xrow + idx));
    }
  }
}

extern "C" void kernel_launch(void* const* d_in, const int* in_sizes, int n_in,
                              void* d_out, int out_size, void* d_ws, size_t ws_size,
                              hipStream_t stream) {
  (void)n_in; (void)out_size; (void)d_ws; (void)ws_size;

  const int*   states    = (const int*)d_in[0];
  const int*   model_ids = (const int*)d_in[1];
  const float* score_t   = (const float*)d_in[2];
  const int*   next_t    = (const int*)d_in[3];
  const float* final_w   = (const float*)d_in[4];
  const float* alphas    = (const float*)d_in[5];
  const int*   eos       = (const int*)d_in[6];

  const int B  = in_sizes[0];
  const int MS = in_sizes[4];          // M*S
  const int M  = in_sizes[5];
  const int S  = MS / M;
  const int V  = in_sizes[2] / MS;

  float* out_scores = (float*)d_out;
  int*   out_next   = (int*)((float*)d_out + (size_t)B * (size_t)V);

  dim3 block(256);
  dim3 grid((V + CHUNK - 1) / CHUNK, B);
  hipLaunchKernelGGL(biasing_multimodel_kernel, grid, block, 0, stream,
                     states, model_ids, score_t, next_t, final_w, alphas, eos,
                     out_scores, out_next, S, V);
}